// MistralAttention_KVmix_32478542692825
// MI455X (gfx1250) — compile-verified
//
#include <hip/hip_runtime.h>
#include <math.h>

#ifndef __has_builtin
#define __has_builtin(x) 0
#endif

#define BB_    4
#define HH_    32
#define KVH_   8
#define DD_    128
#define SS_    8192
#define HID_   4096
#define NREP_  4
#define NKQ_   7168          /* = n_kq = n_vq for cur=8193, ratio .125, GROUP 32 */
#define NSPLIT_ 32
#define CHUNK_  256          /* cache positions per split */
#define NBLK_   8            /* 32-position blocks per split */
#define RSQRT_D_ 0.08838834764831845f

typedef __attribute__((ext_vector_type(2))) float v2f;
typedef __attribute__((ext_vector_type(8))) float v8f;
typedef __attribute__((ext_vector_type(4))) int   v4i;

#if __has_builtin(__builtin_amdgcn_wmma_f32_16x16x4_f32)
#define HAVE_WMMA 1
#else
#define HAVE_WMMA 0
#endif

#if __has_builtin(__builtin_amdgcn_global_load_async_to_lds_b128)
#define HAVE_ASYNC_LDS 1
#else
#define HAVE_ASYNC_LDS 0
#endif

#if HAVE_ASYNC_LDS
typedef __attribute__((address_space(1))) v4i* glob_v4i_ptr;
typedef __attribute__((address_space(3))) v4i* lds_v4i_ptr;
#endif

__device__ __forceinline__ void copy16_to_lds(float* dst_lds, const float* src_glb) {
#if HAVE_ASYNC_LDS
    __builtin_amdgcn_global_load_async_to_lds_b128(
        (glob_v4i_ptr)(float*)src_glb,
        (lds_v4i_ptr)dst_lds, 0, 0);
#else
    *(float4*)dst_lds = *(const float4*)src_glb;
#endif
}

__device__ __forceinline__ void async_wait0() {
#if HAVE_ASYNC_LDS
#if __has_builtin(__builtin_amdgcn_s_wait_asynccnt)
    __builtin_amdgcn_s_wait_asynccnt(0);
#else
    asm volatile("s_wait_asynccnt 0" ::: "memory");
#endif
#endif
}

// ---------------- Kernel A: QKV projection + RoPE --------------------------
// 24 blocks x 256 threads. Each thread owns one output column n across all
// 4 batches (so every weight element is read exactly once, coalesced in n).
__global__ __launch_bounds__(256) void qkv_rope_kernel(
    const float* __restrict__ hidden, const float* __restrict__ wq,
    const float* __restrict__ wk, const float* __restrict__ wv,
    const int* __restrict__ position_ids,
    float* __restrict__ qrope, float* __restrict__ knew, float* __restrict__ vnew)
{
    __shared__ float hs[BB_][1024];
    __shared__ float colbuf[BB_][256];
    const int t = threadIdx.x;
    const int n = blockIdx.x * 256 + t;

    const float* wptr; int ncols, col, region;
    if (n < HID_)                 { wptr = wq; ncols = HID_;     col = n;                    region = 0; }
    else if (n < HID_ + KVH_*DD_) { wptr = wk; ncols = KVH_*DD_; col = n - HID_;             region = 1; }
    else                          { wptr = wv; ncols = KVH_*DD_; col = n - HID_ - KVH_*DD_;  region = 2; }

    float acc[BB_] = {0.f, 0.f, 0.f, 0.f};
    for (int kb = 0; kb < HID_; kb += 1024) {
        __syncthreads();
        for (int i = t; i < BB_*1024; i += 256) {
            int bb = i >> 10, kk = i & 1023;
            hs[bb][kk] = hidden[(size_t)bb*HID_ + kb + kk];
        }
        __syncthreads();
        for (int kk = 0; kk < 1024; ++kk) {
            float w = wptr[(size_t)(kb + kk)*ncols + col];
            acc[0] = fmaf(hs[0][kk], w, acc[0]);
            acc[1] = fmaf(hs[1][kk], w, acc[1]);
            acc[2] = fmaf(hs[2][kk], w, acc[2]);
            acc[3] = fmaf(hs[3][kk], w, acc[3]);
        }
    }

    for (int bb = 0; bb < BB_; ++bb) colbuf[bb][t] = acc[bb];
    __syncthreads();

    const int d  = col & 127;
    const int i2 = d & 63;
    for (int bb = 0; bb < BB_; ++bb) {
        float val = acc[bb];
        float outv;
        if (region == 2) {
            outv = val;
        } else {
            float pos = (float)position_ids[bb];
            float fr  = pos * __powf(10000.f, -(float)i2 * (1.f/64.f));
            float cs  = cosf(fr), sn = sinf(fr);
            float partner = colbuf[bb][t ^ 64];
            float rot = (d < 64) ? -partner : partner;
            outv = val*cs + rot*sn;
        }
        if (region == 0)      qrope[(size_t)bb*HH_*DD_ + n]    = outv;
        else if (region == 1) knew [(size_t)bb*KVH_*DD_ + col] = outv;
        else                  vnew [(size_t)bb*KVH_*DD_ + col] = outv;
    }
}

// ---------------- Kernel B: fused dequant + flash-decode partials ----------
// grid (NSPLIT, KVH, B), 256 threads. Streams K/V via async-to-LDS, dequants
// in LDS, WMMA f32 16x16x4 for the 32pos x 4head score tile, online softmax.
__global__ __launch_bounds__(256) void attn_partial_kernel(
    const float* __restrict__ key_past, const float* __restrict__ value_past,
    const float* __restrict__ qrope, const float* __restrict__ knew,
    const float* __restrict__ vnew, float* __restrict__ partial)
{
    __shared__ float qs[16][DD_];        // rows 0..3 = q heads (pre-scaled), 4..15 = 0
    __shared__ float kbuf[32][DD_];
    __shared__ float vbuf[32][DD_];
    __shared__ float sc[32][NREP_];
    __shared__ float mstate[NREP_], lstate[NREP_], corr4[NREP_];

    const int s = blockIdx.x, g = blockIdx.y, b = blockIdx.z;
    const int t = threadIdx.x;
    const int r0 = t >> 7;           // this thread accumulates heads r0 and r0+2
    const int dd = t & 127;

    for (int i = t; i < 16*DD_; i += 256) {
        int r = i >> 7, d = i & 127;
        qs[r][d] = (r < NREP_)
            ? qrope[((size_t)b*HH_ + (g*NREP_ + r))*DD_ + d] * RSQRT_D_
            : 0.f;
    }
    if (t < NREP_) { mstate[t] = -INFINITY; lstate[t] = 0.f; }
    float acc0 = 0.f, acc1 = 0.f;
    __syncthreads();

#if HAVE_WMMA
    // Hoist the loop-invariant q-side (B-matrix) WMMA fragments into registers.
    const int wave = t >> 5;
    const int lane = t & 31;
    const int mrow = lane & 15;
    const int hi   = lane >> 4;
    v2f bfrag[32];
    if (wave < 2) {
        for (int kc = 0; kc < 32; ++kc) {
            const int ka = kc*4 + (hi << 1);
            bfrag[kc].x = qs[mrow][ka];
            bfrag[kc].y = qs[mrow][ka + 1];
        }
    }
#endif

    const size_t kvbase = ((size_t)b*KVH_ + g) * (size_t)SS_ * DD_;

    for (int j = 0; j < NBLK_; ++j) {
        const int gpos = s*CHUNK_ + j*32;
        const bool quant = (gpos < NKQ_);
        const float* kp = key_past   + kvbase + (size_t)gpos*DD_;
        const float* vp = value_past + kvbase + (size_t)gpos*DD_;
        for (int it = 0; it < 4; ++it) {
            int idx4 = t + it*256;              // 1024 x float4 per buffer
            int row = idx4 >> 5, c4 = (idx4 & 31) << 2;
            copy16_to_lds(&kbuf[row][c4], kp + row*DD_ + c4);
            copy16_to_lds(&vbuf[row][c4], vp + row*DD_ + c4);
        }
        async_wait0();
        __syncthreads();

        if (quant && t < 128) {
            // K fake-quant: group = 32 consecutive positions per channel d=t
            {
                float mn = kbuf[0][t], mx = mn;
                for (int i = 1; i < 32; ++i) {
                    float v = kbuf[i][t]; mn = fminf(mn, v); mx = fmaxf(mx, v);
                }
                float scale = (mx - mn) * (1.f/3.f);
                float inv = (scale > 0.f) ? (1.f/scale) : 0.f;
                for (int i = 0; i < 32; ++i) {
                    float v = kbuf[i][t];
                    float qv = rintf((v - mn) * inv);
                    qv = fminf(fmaxf(qv, 0.f), 3.f);
                    kbuf[i][t] = fmaf(qv, scale, mn);
                }
            }
            // V fake-quant: group = 32 consecutive channels per position
            {
                int row = t >> 2, d0 = (t & 3) << 5;
                float mn = vbuf[row][d0], mx = mn;
                for (int i = 1; i < 32; ++i) {
                    float v = vbuf[row][d0 + i]; mn = fminf(mn, v); mx = fmaxf(mx, v);
                }
                float scale = (mx - mn) * (1.f/3.f);
                float inv = (scale > 0.f) ? (1.f/scale) : 0.f;
                for (int i = 0; i < 32; ++i) {
                    float v = vbuf[row][d0 + i];
                    float qv = rintf((v - mn) * inv);
                    qv = fminf(fmaxf(qv, 0.f), 3.f);
                    vbuf[row][d0 + i] = fmaf(qv, scale, mn);
                }
            }
        }
        __syncthreads();

        // Scores: sc[pos][r] = (K_dq[pos,:] . q_scaled[r,:])
#if HAVE_WMMA
        if (wave < 2) {                       // wave-uniform: EXEC stays full
            const int tile = wave;            // positions [tile*16, tile*16+16)
            v8f c = {0.f,0.f,0.f,0.f,0.f,0.f,0.f,0.f};
            for (int kc = 0; kc < 32; ++kc) {
                const int ka = kc*4 + (hi << 1);
                v2f a;
                a.x = kbuf[tile*16 + mrow][ka];
                a.y = kbuf[tile*16 + mrow][ka + 1];
                c = __builtin_amdgcn_wmma_f32_16x16x4_f32(
                        false, a, false, bfrag[kc], (short)0, c, false, false);
            }
            for (int jj = 0; jj < 8; ++jj) {
                int pos = tile*16 + jj + (hi << 3);
                if (mrow < NREP_) sc[pos][mrow] = c[jj];
            }
        }
#else
        if (t < 128) {
            int pos = t >> 2, r = t & 3;
            float sdot = 0.f;
            for (int d2 = 0; d2 < DD_; ++d2) sdot = fmaf(kbuf[pos][d2], qs[r][d2], sdot);
            sc[pos][r] = sdot;
        }
#endif
        __syncthreads();

        if (t < NREP_) {
            float bm = -INFINITY;
            for (int i = 0; i < 32; ++i) bm = fmaxf(bm, sc[i][t]);
            float mold = mstate[t];
            float mnew = fmaxf(mold, bm);
            float c = __expf(mold - mnew);
            float add = 0.f;
            for (int i = 0; i < 32; ++i) {
                float p = __expf(sc[i][t] - mnew);
                sc[i][t] = p; add += p;
            }
            lstate[t] = lstate[t]*c + add;
            mstate[t] = mnew;
            corr4[t]  = c;
        }
        __syncthreads();

        acc0 *= corr4[r0];
        acc1 *= corr4[r0 + 2];
        for (int i = 0; i < 32; ++i) {
            acc0 = fmaf(sc[i][r0],     vbuf[i][dd], acc0);
            acc1 = fmaf(sc[i][r0 + 2], vbuf[i][dd], acc1);
        }
        __syncthreads();
    }

    // Last split also folds in the freshly projected (RoPE'd) token at pos S
    if (s == NSPLIT_ - 1) {
        if (t < DD_) {
            kbuf[0][t] = knew[((size_t)b*KVH_ + g)*DD_ + t];
            vbuf[0][t] = vnew[((size_t)b*KVH_ + g)*DD_ + t];
        }
        __syncthreads();
        if (t < NREP_) {
            float sdot = 0.f;
            for (int d2 = 0; d2 < DD_; ++d2) sdot = fmaf(qs[t][d2], kbuf[0][d2], sdot);
            float mold = mstate[t];
            float mnew = fmaxf(mold, sdot);
            float c = __expf(mold - mnew);
            float p = __expf(sdot - mnew);
            lstate[t] = lstate[t]*c + p;
            mstate[t] = mnew;
            corr4[t]  = c;
            sc[0][t]  = p;
        }
        __syncthreads();
        acc0 = fmaf(sc[0][r0],     vbuf[0][dd], acc0 * corr4[r0]);
        acc1 = fmaf(sc[0][r0 + 2], vbuf[0][dd], acc1 * corr4[r0 + 2]);
        __syncthreads();
    }

    // partial layout: [b*H + h][split] -> {m, l, acc[128]}
    {
        float* p0 = partial + (((size_t)(b*HH_ + g*NREP_ + r0)     * NSPLIT_) + s) * (DD_ + 2);
        float* p1 = partial + (((size_t)(b*HH_ + g*NREP_ + r0 + 2) * NSPLIT_) + s) * (DD_ + 2);
        p0[2 + dd] = acc0;
        p1[2 + dd] = acc1;
        if (t < NREP_) {
            float* pr = partial + (((size_t)(b*HH_ + g*NREP_ + t) * NSPLIT_) + s) * (DD_ + 2);
            pr[0] = mstate[t];
            pr[1] = lstate[t];
        }
    }
}

// ---------------- Kernel C: split reduction --------------------------------
__global__ __launch_bounds__(128) void combine_kernel(
    const float* __restrict__ partial, float* __restrict__ attn)
{
    const int hflat = blockIdx.x;       // b*H + h
    const int d = threadIdx.x;
    const float* pb = partial + (size_t)hflat * NSPLIT_ * (DD_ + 2);
    float M = -INFINITY;
    for (int s2 = 0; s2 < NSPLIT_; ++s2) M = fmaxf(M, pb[s2*(DD_ + 2)]);
    float L = 0.f, o = 0.f;
    for (int s2 = 0; s2 < NSPLIT_; ++s2) {
        float w = __expf(pb[s2*(DD_ + 2)] - M);
        L += pb[s2*(DD_ + 2) + 1] * w;
        o  = fmaf(w, pb[s2*(DD_ + 2) + 2 + d], o);
    }
    attn[(size_t)hflat*DD_ + d] = o / L;
}

// ---------------- Kernel D: output projection ------------------------------
__global__ __launch_bounds__(256) void out_proj_kernel(
    const float* __restrict__ attn, const float* __restrict__ wo,
    float* __restrict__ out)
{
    __shared__ float abuf[BB_][1024];
    const int t = threadIdx.x;
    const int n = blockIdx.x * 256 + t;
    float acc[BB_] = {0.f, 0.f, 0.f, 0.f};
    for (int kb = 0; kb < HID_; kb += 1024) {
        __syncthreads();
        for (int i = t; i < BB_*1024; i += 256) {
            int bb = i >> 10, kk = i & 1023;
            abuf[bb][kk] = attn[(size_t)bb*HID_ + kb + kk];
        }
        __syncthreads();
        for (int kk = 0; kk < 1024; ++kk) {
            float w = wo[(size_t)(kb + kk)*HID_ + n];
            acc[0] = fmaf(abuf[0][kk], w, acc[0]);
            acc[1] = fmaf(abuf[1][kk], w, acc[1]);
            acc[2] = fmaf(abuf[2][kk], w, acc[2]);
            acc[3] = fmaf(abuf[3][kk], w, acc[3]);
        }
    }
    for (int bb = 0; bb < BB_; ++bb) out[(size_t)bb*HID_ + n] = acc[bb];
}

// ---------------------------------------------------------------------------
extern "C" void kernel_launch(void* const* d_in, const int* in_sizes, int n_in,
                              void* d_out, int out_size, void* d_ws, size_t ws_size,
                              hipStream_t stream) {
    (void)in_sizes; (void)n_in; (void)out_size; (void)ws_size;
    const float* hidden       = (const float*)d_in[0];
    const float* key_past     = (const float*)d_in[1];
    const float* value_past   = (const float*)d_in[2];
    const float* wq           = (const float*)d_in[3];
    const float* wk           = (const float*)d_in[4];
    const float* wv           = (const float*)d_in[5];
    const float* wo           = (const float*)d_in[6];
    const int*   position_ids = (const int*)d_in[7];
    float* out = (float*)d_out;

    float* ws      = (float*)d_ws;
    float* qrope   = ws;                                      // B*H*D
    float* knew    = qrope + (size_t)BB_*HH_*DD_;             // B*KVH*D
    float* vnew    = knew  + (size_t)BB_*KVH_*DD_;            // B*KVH*D
    float* partial = vnew  + (size_t)BB_*KVH_*DD_;            // B*H*NSPLIT*(D+2)
    float* attn    = partial + (size_t)BB_*HH_*NSPLIT_*(DD_ + 2); // B*H*D

    qkv_rope_kernel<<<(HID_ + 2*KVH_*DD_) / 256, 256, 0, stream>>>(
        hidden, wq, wk, wv, position_ids, qrope, knew, vnew);
    attn_partial_kernel<<<dim3(NSPLIT_, KVH_, BB_), 256, 0, stream>>>(
        key_past, value_past, qrope, knew, vnew, partial);
    combine_kernel<<<BB_*HH_, 128, 0, stream>>>(partial, attn);
    out_proj_kernel<<<HID_ / 256, 256, 0, stream>>>(attn, wo, out);
}